// LocalBinaryLayer_13537736917574
// MI455X (gfx1250) — compile-verified
//
#include <hip/hip_runtime.h>
#include <stdint.h>

// Problem constants (match reference: x is (32,3,512,512) fp32, 8 bins, density)
#define IMGS 96            // 32 * 3 planes
#define IMG_H 512
#define IMG_W 512
#define STRIP_ROWS 16
#define NSTRIPS (IMG_H / STRIP_ROWS)   // 32
#define TROWS (STRIP_ROWS + 1)         // 17 rows: centers + one below-halo
#define NBINS 8

// Bilinear weights for radius-1 diagonal samples (t = 1 - sqrt(2)/2, etc.)
#define WA 0.20710678f     // 0.29289322*0.70710678
#define WS 0.08578644f     // 0.29289322^2
// density norm: 1 / (H*W * (255/8)) = 1 / 8,355,840
#define NORM (1.0f / 8355840.0f)

typedef unsigned int uint32x4 __attribute__((ext_vector_type(4)));
typedef int int32x4 __attribute__((ext_vector_type(4)));
typedef int int32x8 __attribute__((ext_vector_type(8)));

__global__ __launch_bounds__(256) void lbp_zero_out(float* out) {
    int i = blockIdx.x * 256 + threadIdx.x;
    if (i < IMGS * NBINS) out[i] = 0.0f;
}

__global__ __launch_bounds__(256) void lbp_finalize(float* out) {
    int i = blockIdx.x * 256 + threadIdx.x;
    if (i < IMGS * NBINS) out[i] = out[i] * NORM;
}

__global__ __launch_bounds__(256) void lbp_main(const float* __restrict__ x,
                                                float* __restrict__ out) {
    __shared__ float    tile[TROWS * IMG_W];     // 34,816 B
    __shared__ unsigned lanesum[2 * 256];        //  2,048 B
    __shared__ unsigned partsum[64];

    const int tid   = threadIdx.x;
    const int strip = blockIdx.x;
    const int img   = blockIdx.y;
    const int r0    = strip * STRIP_ROWS;

    // ------------------------------------------------------------------
    // Stage rows [r0 .. r0+16] of this plane into LDS via the Tensor
    // Data Mover (one DMA per block; bottom halo past the image reads
    // as zero via TDM out-of-bounds semantics).
    // ------------------------------------------------------------------
#if __has_builtin(__builtin_amdgcn_tensor_load_to_lds)
    if (tid < 32) {  // TDM is a per-wave op (EXEC ignored) -> only wave 0 issues it
        const uint64_t gaddr =
            (uint64_t)(uintptr_t)(x + (size_t)img * (IMG_H * IMG_W) + (size_t)r0 * IMG_W);
        const unsigned lds_base = (unsigned)(uintptr_t)&tile[0];
        const unsigned tdim1    = (unsigned)(IMG_H - r0);  // rows available; rows past
                                                           // this read as zero (TDM OOB)
        uint32x4 g0;
        g0.x = 1u;                                   // count=1 (valid), user mode
        g0.y = lds_base;                             // lds_addr (bytes)
        g0.z = (unsigned)gaddr;                      // global_addr[31:0]
        g0.w = (unsigned)((gaddr >> 32) & 0x01FFFFFFu) | (2u << 30); // addr[56:32]|type=2

        int32x8 g1;
        g1[0] = 0x00020000;                          // data_size=2 (4B), mask=0, no pad
        g1[1] = (int)(((unsigned)IMG_W & 0xFFFFu) << 16);            // tensor_dim0 lo16
        g1[2] = (int)((((unsigned)IMG_W >> 16) & 0xFFFFu) |
                      ((tdim1 & 0xFFFFu) << 16));                    // dim0 hi / dim1 lo
        g1[3] = (int)(((tdim1 >> 16) & 0xFFFFu) |
                      (((unsigned)IMG_W & 0xFFFFu) << 16));          // dim1 hi / tile_dim0
        g1[4] = (int)(TROWS & 0xFFFFu);              // tile_dim1=17, tile_dim2=0
        g1[5] = (int)IMG_W;                          // tensor_dim0_stride lo32
        g1[6] = 0;                                   // stride hi16, dim1_stride lo16
        g1[7] = 0;
        int32x4 gz4 = {0, 0, 0, 0};
        int32x8 gz8 = {0, 0, 0, 0, 0, 0, 0, 0};
        __builtin_amdgcn_tensor_load_to_lds(g0, g1, gz4, gz4, gz8, 0);
        __builtin_amdgcn_s_wait_tensorcnt(0);
    }
#else
    // Fallback staging: coalesced global loads -> LDS stores
    {
        const size_t base = (size_t)img * (IMG_H * IMG_W);
        for (int i = tid; i < TROWS * IMG_W; i += 256) {
            const int rr = i >> 9;          // / 512
            const int cc = i & (IMG_W - 1);
            const int gr = r0 + rr;
            tile[i] = (gr < IMG_H) ? x[base + (size_t)gr * IMG_W + cc] : 0.0f;
        }
    }
#endif
    __syncthreads();

    // ------------------------------------------------------------------
    // LBP + binning. bin = code >> 5 depends only on bits 5,6,7:
    //   b5: bilinear at (+0.7071,-0.7071), b6: (r+1,c), b7: (+0.7071,+0.7071)
    // 2x3 sliding window walking down each column.
    // ------------------------------------------------------------------
    unsigned cl = 0u, ch = 0u;   // 8 per-lane byte counters (<=32 each)

    #pragma unroll
    for (int hx = 0; hx < 2; ++hx) {
        const int   xx  = tid + hx * 256;
        const int   xm1 = (xx == 0)         ? 0         : xx - 1;
        const int   xp1 = (xx == IMG_W - 1) ? IMG_W - 1 : xx + 1;
        const float mL  = (xx == 0)         ? 0.0f : 1.0f;  // zero-padded border
        const float mR  = (xx == IMG_W - 1) ? 0.0f : 1.0f;

        float pc0 = tile[xm1] * mL;
        float pc1 = tile[xx];
        float pc2 = tile[xp1] * mR;

        #pragma unroll
        for (int r = 1; r <= STRIP_ROWS; ++r) {
            const int   o   = r * IMG_W;
            const float pb0 = tile[o + xm1] * mL;
            const float pb1 = tile[o + xx];
            const float pb2 = tile[o + xp1] * mR;

            const float c  = pc1;
            const float sc = WS * c;
            const float s5 = fmaf(0.5f, pb0, fmaf(WA, pc0 + pb1, sc));
            const float s7 = fmaf(0.5f, pb2, fmaf(WA, pc2 + pb1, sc));

            const unsigned bin = (s5  >= c ? 1u : 0u)
                               | (pb1 >= c ? 2u : 0u)
                               | (s7  >= c ? 4u : 0u);

            const unsigned inc = 1u << ((bin & 3u) << 3);
            cl += (bin & 4u) ? 0u  : inc;
            ch += (bin & 4u) ? inc : 0u;

            pc0 = pb0; pc1 = pb1; pc2 = pb2;
        }
    }

    // ------------------------------------------------------------------
    // Block reduction of 256 lanes x 8 byte-counters -> 8 bin counts
    // ------------------------------------------------------------------
    lanesum[tid]       = cl;
    lanesum[256 + tid] = ch;
    __syncthreads();

    if (tid < 64) {
        const unsigned bin = (unsigned)tid >> 3;
        const unsigned j0  = ((unsigned)tid & 7u) * 32u;
        const unsigned d   = (bin >> 2) * 256u;
        const unsigned sh  = (bin & 3u) * 8u;
        unsigned s = 0u;
        #pragma unroll
        for (unsigned j = 0; j < 32u; ++j)
            s += (lanesum[d + j0 + j] >> sh) & 0xFFu;
        partsum[tid] = s;
    }
    __syncthreads();

    if (tid < NBINS) {
        unsigned tot = 0u;
        #pragma unroll
        for (int k = 0; k < 8; ++k) tot += partsum[tid * 8 + k];
        // counts <= 262144: exact in fp32, order-independent -> deterministic
        atomicAdd(&out[img * NBINS + tid], (float)tot);
    }
}

extern "C" void kernel_launch(void* const* d_in, const int* in_sizes, int n_in,
                              void* d_out, int out_size, void* d_ws, size_t ws_size,
                              hipStream_t stream) {
    (void)in_sizes; (void)n_in; (void)d_ws; (void)ws_size; (void)out_size;
    const float* x  = (const float*)d_in[0];
    float*       out = (float*)d_out;

    lbp_zero_out<<<dim3(3), dim3(256), 0, stream>>>(out);
    lbp_main<<<dim3(NSTRIPS, IMGS), dim3(256), 0, stream>>>(x, out);
    lbp_finalize<<<dim3(3), dim3(256), 0, stream>>>(out);
}